// NUFFTLayerMultiChannelInit_59906203845044
// MI455X (gfx1250) — compile-verified
//
#include <hip/hip_runtime.h>
#include <math.h>

// ---------------- problem constants ----------------
#define NM     4097            // mesh points (odd, = SCALE since L = 2*pi)
#define NPAD   4112            // 257 * 16 : padded mesh for 16-wide WMMA tiling
#define JT     257             // NPAD / 16 column tiles
#define BATCH  32
#define NP     256
constexpr double LBOX_D  = 6.283185307179586476925286766559;   // 2*pi
constexpr double TAU_D   = 12.0 / ((double)NM * (double)NM);
constexpr float  LBOX    = (float)LBOX_D;
constexpr float  DX      = (float)(LBOX_D / (double)NM);        // grid spacing = 2*pi/N
constexpr float  INVDX   = (float)((double)NM / LBOX_D);
constexpr float  INV4TAU = (float)(1.0 / (4.0 * TAU_D));
constexpr float  TAU_F   = (float)TAU_D;
constexpr float  INVSC   = (float)(1.0 / (double)NM);           // 1/SCALE = 1/N

typedef __attribute__((ext_vector_type(2))) float v2f;
typedef __attribute__((ext_vector_type(8))) float v8f;
struct __align__(8) f2 { float x, y; };    // (cos, -sin) pair -> ds_load_b64

// ---------------- stage 1: gridding (dense, deterministic gather) ----------------
// rho_t[n][b] (transposed so WMMA A-operand loads are lane-coalesced)
__global__ __launch_bounds__(256) void grid_rho(const float* __restrict__ x,
                                                float* __restrict__ rho_t) {
  int tid = blockIdx.x * 256 + threadIdx.x;
  if (tid >= BATCH * NPAD) return;
  int b = tid & 31;
  int n = tid >> 5;
  if (n >= NM) { rho_t[n * BATCH + b] = 0.0f; return; }   // zero K-padding
  float xg = (float)n * DX;
  const float* xr = x + b * NP;
  float acc = 0.0f;
  for (int p = 0; p < NP; ++p) {
    float d  = xr[p] - xg;
    float dm = d - LBOX, dp = d + LBOX;
    acc += __expf(-d  * d  * INV4TAU)
         + __expf(-dm * dm * INV4TAU)
         + __expf(-dp * dp * INV4TAU);
  }
  rho_t[n * BATCH + b] = acc;
}

// ---------------- stage 2a: forward DFT as f32 WMMA GEMM ----------------
// F_re[b][j] = sum_n rho[b,n] cos(2*pi*(j-2048)*n/N)
// F_im[b][j] = sum_n rho[b,n] * (-sin(...))
// One wave computes a 16(b) x 16(j) tile of both via V_WMMA_F32_16X16X4_F32.
__global__ __launch_bounds__(256) void fwd_dft(const float* __restrict__ rho_t,
                                               float* __restrict__ Fre,
                                               float* __restrict__ Fim) {
  __shared__ f2 tw[NM];                                   // 32 KB twiddle table in LDS
  for (int i = threadIdx.x; i < NM; i += 256) {
    float sv, cv;
    __sincosf((float)i * DX, &sv, &cv);                   // 2*pi*i/N  (DX == 2*pi/N)
    tw[i].x = cv; tw[i].y = -sv;
  }
  __syncthreads();

  int wave = blockIdx.x * 8 + (threadIdx.x >> 5);
  if (wave >= 2 * JT) return;
  int lane = threadIdx.x & 31;
  int bb = wave & 1, jb = wave >> 1;
  int colj = jb * 16 + (lane & 15);                       // output column j
  int s = (colj - 2048) % NM; if (s < 0) s += NM;         // frequency multiplier mod N
  int dl = (lane >= 16) ? 2 : 0;                          // K-offset of this lane half
  int m0   = (dl * s) % NM;                               // angle index for A/B VGPR0
  int m1   = ((dl + 1) * s) % NM;                         // ... VGPR1
  int step = (4 * s) % NM;                                // K advances by 4 per iter
  int rowA = bb * 16 + (lane & 15);                       // A row = batch index

  v8f acc_re = {0,0,0,0,0,0,0,0};
  v8f acc_im = {0,0,0,0,0,0,0,0};
  for (int k0 = 0; k0 < NPAD; k0 += 4) {
    v2f a;
    a.x = rho_t[(k0 + dl)     * BATCH + rowA];
    a.y = rho_t[(k0 + dl + 1) * BATCH + rowA];
    __builtin_prefetch(rho_t + (k0 + 4 + dl) * BATCH + rowA, 0, 1);
    f2 t0 = tw[m0], t1 = tw[m1];
    v2f bc; bc.x = t0.x; bc.y = t1.x;                     // cos rows
    v2f bs; bs.x = t0.y; bs.y = t1.y;                     // -sin rows
    acc_re = __builtin_amdgcn_wmma_f32_16x16x4_f32(false, a, false, bc,
                                                   (short)0, acc_re, false, false);
    acc_im = __builtin_amdgcn_wmma_f32_16x16x4_f32(false, a, false, bs,
                                                   (short)0, acc_im, false, false);
    m0 += step; if (m0 >= NM) m0 -= NM;
    m1 += step; if (m1 >= NM) m1 -= NM;
  }
  int rbase = bb * 16 + ((lane >= 16) ? 8 : 0);           // C/D layout: VGPR v -> M=v | v+8
  for (int v = 0; v < 8; ++v) {
    Fre[(rbase + v) * NPAD + colj] = acc_re[v];
    Fim[(rbase + v) * NPAD + colj] = acc_im[v];
  }
}

// ---------------- stage 2b: spectral pointwise (deconv + mult_re/mult_im) ----------------
// P_t[j][r=2b+c] = mult_re*(re-im)*deconv ; Q_t = mult_im*(re+im)*deconv
__global__ __launch_bounds__(256) void spectral(const float* __restrict__ Fre,
                                                const float* __restrict__ Fim,
                                                const float* __restrict__ mre,
                                                const float* __restrict__ mim,
                                                float* __restrict__ Pt,
                                                float* __restrict__ Qt) {
  int tid = blockIdx.x * 256 + threadIdx.x;
  if (tid >= BATCH * NPAD) return;
  int j = tid % NPAD;
  int b = tid / NPAD;
  if (j >= NM) {                                          // zero K-padding rows
    Pt[j * 64 + b * 2 + 0] = 0.0f; Pt[j * 64 + b * 2 + 1] = 0.0f;
    Qt[j * 64 + b * 2 + 0] = 0.0f; Qt[j * 64 + b * 2 + 1] = 0.0f;
    return;
  }
  float kj = (float)(j - 2048);
  float dcoef = sqrtf((float)(3.14159265358979323846 / TAU_D));
  float d  = dcoef * __expf(kj * kj * TAU_F);             // gaussian_deconv(k, tau)
  float re = Fre[b * NPAD + j] * d * INVSC;
  float im = Fim[b * NPAD + j] * d * INVSC;
  for (int c = 0; c < 2; ++c) {
    float rp = mre[c * NM + j] * (re - im);
    float ip = mim[c * NM + j] * (re + im);               // faithful (mult_im is zeros)
    Pt[j * 64 + b * 2 + c] = rp * d;
    Qt[j * 64 + b * 2 + c] = ip * d;
  }
}

// ---------------- stage 3: inverse DFT as f32 WMMA GEMM ----------------
// irf[r][n] = (1/N) * sum_j ( P[r,j]*cos(th) + Q[r,j]*(-sin(th)) ), th = 2*pi*(j-2048)*n/N
__global__ __launch_bounds__(256) void inv_dft(const float* __restrict__ Pt,
                                               const float* __restrict__ Qt,
                                               float* __restrict__ irf) {
  __shared__ f2 tw[NM];
  for (int i = threadIdx.x; i < NM; i += 256) {
    float sv, cv;
    __sincosf((float)i * DX, &sv, &cv);
    tw[i].x = cv; tw[i].y = -sv;
  }
  __syncthreads();

  int wave = blockIdx.x * 8 + (threadIdx.x >> 5);
  if (wave >= 4 * JT) return;
  int lane = threadIdx.x & 31;
  int rb = wave & 3, nb = wave >> 2;
  int coln = nb * 16 + (lane & 15);                       // output column n
  int u = coln % NM;                                      // multiplier mod N
  int dl = (lane >= 16) ? 2 : 0;
  int m0   = ((dl     + NM - 2048) * u) % NM;             // <= 2052*4111 fits int32
  int m1   = ((dl + 1 + NM - 2048) * u) % NM;
  int step = (4 * u) % NM;
  int rowA = rb * 16 + (lane & 15);                       // A row = 2b+c

  v8f acc = {0,0,0,0,0,0,0,0};
  for (int j0 = 0; j0 < NPAD; j0 += 4) {
    v2f p, q;
    p.x = Pt[(j0 + dl)     * 64 + rowA];
    p.y = Pt[(j0 + dl + 1) * 64 + rowA];
    q.x = Qt[(j0 + dl)     * 64 + rowA];
    q.y = Qt[(j0 + dl + 1) * 64 + rowA];
    __builtin_prefetch(Pt + (j0 + 4 + dl) * 64 + rowA, 0, 1);
    f2 t0 = tw[m0], t1 = tw[m1];
    v2f bc; bc.x = t0.x; bc.y = t1.x;
    v2f bs; bs.x = t0.y; bs.y = t1.y;
    acc = __builtin_amdgcn_wmma_f32_16x16x4_f32(false, p, false, bc,
                                                (short)0, acc, false, false);
    acc = __builtin_amdgcn_wmma_f32_16x16x4_f32(false, q, false, bs,
                                                (short)0, acc, false, false);
    m0 += step; if (m0 >= NM) m0 -= NM;
    m1 += step; if (m1 >= NM) m1 -= NM;
  }
  int rbase = rb * 16 + ((lane >= 16) ? 8 : 0);
  for (int v = 0; v < 8; ++v)
    irf[(rbase + v) * NPAD + coln] = acc[v] * INVSC;      // ifft 1/N
}

// ---------------- stage 4: interpolation back to particles ----------------
// Gaussian support: exp(-0.8225*m^2) -> exact zero in fp32 beyond |m|=12; W=16 is exact.
__global__ __launch_bounds__(256) void interp(const float* __restrict__ x,
                                              const float* __restrict__ irf,
                                              float* __restrict__ out) {
  int tid = blockIdx.x * 256 + threadIdx.x;
  if (tid >= BATCH * NP) return;
  int b = tid / NP;
  float xv = x[tid];
  int i0 = (int)floorf(xv * INVDX + 0.5f);
  const float* r0 = irf + (b * 2 + 0) * NPAD;
  const float* r1 = irf + (b * 2 + 1) * NPAD;
  float a0 = 0.0f, a1 = 0.0f;
  for (int m = -16; m <= 16; ++m) {
    int n = i0 + m;
    n = (n < 0) ? n + NM : ((n >= NM) ? n - NM : n);
    float d  = xv - (float)n * DX;
    float dm = d - LBOX, dp = d + LBOX;
    float g = __expf(-d  * d  * INV4TAU)
            + __expf(-dm * dm * INV4TAU)
            + __expf(-dp * dp * INV4TAU);
    a0 += g * r0[n];
    a1 += g * r1[n];
  }
  out[tid * 2 + 0] = a0 * INVSC;                          // /SCALE
  out[tid * 2 + 1] = a1 * INVSC;
}

// ---------------- launcher ----------------
extern "C" void kernel_launch(void* const* d_in, const int* in_sizes, int n_in,
                              void* d_out, int out_size, void* d_ws, size_t ws_size,
                              hipStream_t stream) {
  const float* x   = (const float*)d_in[0];   // (32, 256)
  const float* mre = (const float*)d_in[1];   // (2, 4097)
  const float* mim = (const float*)d_in[2];   // (2, 4097)
  float* out = (float*)d_out;                 // (32, 256, 2)

  float* ws    = (float*)d_ws;                // ~4.8 MB total, all written before read
  float* rho_t = ws;  ws += (size_t)NPAD * BATCH;   // [NPAD][32]
  float* Fre   = ws;  ws += (size_t)BATCH * NPAD;   // [32][NPAD]
  float* Fim   = ws;  ws += (size_t)BATCH * NPAD;   // [32][NPAD]
  float* Pt    = ws;  ws += (size_t)NPAD * 64;      // [NPAD][64]
  float* Qt    = ws;  ws += (size_t)NPAD * 64;      // [NPAD][64]
  float* irf   = ws;                                 // [64][NPAD]

  grid_rho<<<(BATCH * NPAD + 255) / 256, 256, 0, stream>>>(x, rho_t);
  fwd_dft <<<(2 * JT + 7) / 8,          256, 0, stream>>>(rho_t, Fre, Fim);
  spectral<<<(BATCH * NPAD + 255) / 256, 256, 0, stream>>>(Fre, Fim, mre, mim, Pt, Qt);
  inv_dft <<<(4 * JT + 7) / 8,          256, 0, stream>>>(Pt, Qt, irf);
  interp  <<<(BATCH * NP + 255) / 256,  256, 0, stream>>>(x, irf, out);
}